// NeuronLevelModels_79293686218889
// MI455X (gfx1250) — compile-verified
//
#include <hip/hip_runtime.h>
#include <hip/hip_bf16.h>
#include <math.h>

typedef __attribute__((ext_vector_type(2))) float v2f;
typedef __attribute__((ext_vector_type(8))) float v8f;

#define B_SZ 512
#define D_SZ 4096
#define M_SZ 25
#define H_SZ 32
#define HB_STRIDE 34     // hbuf row stride (even -> 8B-aligned pairs, merges to ds_load_b64)

// Branch-free erf, Abramowitz-Stegun 7.1.26 (max abs error ~1.5e-7).
// Uses native v_rcp_f32 / v_exp_f32 (exp2); no divergence, no libm call.
__device__ __forceinline__ float erf_fast(float x) {
    const float ax = __builtin_fabsf(x);
    const float t  = __builtin_amdgcn_rcpf(__builtin_fmaf(0.3275911f, ax, 1.0f));
    float p = __builtin_fmaf(t, 1.061405429f, -1.453152027f);
    p = __builtin_fmaf(t, p, 1.421413741f);
    p = __builtin_fmaf(t, p, -0.284496736f);
    p = __builtin_fmaf(t, p, 0.254829592f);
    // exp(-ax*ax) == exp2(-ax*ax * log2(e)) via native v_exp_f32
    const float e = __builtin_amdgcn_exp2f(-ax * ax * 1.4426950408889634f);
    float r = __builtin_fmaf(-t * p, e, 1.0f);
    // copysign(r, x) via bit ops (branch-free)
    unsigned int ur = (__builtin_bit_cast(unsigned int, r) & 0x7fffffffu) |
                      (__builtin_bit_cast(unsigned int, x) & 0x80000000u);
    return __builtin_bit_cast(float, ur);
}

__device__ __forceinline__ float gelu_erf(float x) {
    // torch nn.GELU default: 0.5*x*(1+erf(x/sqrt(2)))
    return 0.5f * x * (1.0f + erf_fast(x * 0.70710678118654752f));
}

__global__ __launch_bounds__(256)
void neuron_mlp_kernel(const float* __restrict__ history,  // [B, D, M]
                       const float* __restrict__ W1,       // [D, H, M]
                       const float* __restrict__ b1,       // [D, H]
                       const float* __restrict__ Wm,       // [1, D, H, H]
                       const float* __restrict__ bm,       // [1, D, H]
                       const float* __restrict__ W2,       // [D, H]
                       const float* __restrict__ b2,       // [D]
                       float* __restrict__ out) {          // [B, D]
    // Weight fragments stored in WMMA per-lane order: one aligned v2f per (k-step, n-tile, lane).
    __shared__ v2f  fB1[14 * 32];               // GEMM1: 7 k-steps x 2 n-tiles x 32 lanes
    __shared__ v2f  fB2[16 * 32];               // GEMM2: 8 k-steps x 2 n-tiles x 32 lanes
    __shared__ float sW2[H_SZ];
    __shared__ float sB1[H_SZ];
    __shared__ float sBm[H_SZ];
    __shared__ float sHbuf[8][16 * HB_STRIDE];  // per-wave activation staging (C-layout -> A-layout bridge)

    const int d    = blockIdx.x;
    const int tid  = threadIdx.x;
    const int wave = tid >> 5;
    const int lane = tid & 31;
    const int lhalf = lane >> 4;   // 0 for lanes 0-15, 1 for lanes 16-31
    const int lidx  = lane & 15;

    // ---- build per-lane weight fragments directly from global (once per block) ----
    // entry e -> (frag = e/32, lane = e%32); frag = s*2 + nt
    for (int e = tid; e < 14 * 32; e += 256) {
        const int fr = e >> 5, ln = e & 31;
        const int s  = fr >> 1, nt = fr & 1;
        const int n  = (ln & 15) + 16 * nt;
        const int kb = 4 * s + 2 * (ln >> 4);
        const float* w = W1 + ((size_t)d * H_SZ + n) * M_SZ;
        v2f f;
        f.x = (kb     < M_SZ) ? w[kb]     : 0.0f;  // zero-pad K 25..27 (setup-only branch)
        f.y = (kb + 1 < M_SZ) ? w[kb + 1] : 0.0f;
        fB1[e] = f;
    }
    for (int e = tid; e < 16 * 32; e += 256) {
        const int fr = e >> 5, ln = e & 31;
        const int s  = fr >> 1, nt = fr & 1;
        const int n  = (ln & 15) + 16 * nt;
        const int kb = 4 * s + 2 * (ln >> 4);
        // out[g] = sum_h in[h]*Wm[d,g,h] => Bmat[k=h][n=g] = Wm[d, n, k]
        const float* w = Wm + ((size_t)d * H_SZ + n) * H_SZ;
        v2f f;
        f.x = w[kb];
        f.y = w[kb + 1];
        fB2[e] = f;
    }
    if (tid < H_SZ) {
        sW2[tid] = W2[(size_t)d * H_SZ + tid];
        sB1[tid] = b1[(size_t)d * H_SZ + tid];
        sBm[tid] = bm[(size_t)d * H_SZ + tid];
    }
    __syncthreads();

    const float b2d   = b2[d];
    const float lmask = (lhalf == 0) ? 1.0f : 0.0f;   // masks the k=24 tail for the high half-wave

    // ---- hoist loop-invariant operands into registers (single ds_load_b64 each) ----
    v2f wB1a[7], wB1b[7];
    #pragma unroll
    for (int s = 0; s < 7; ++s) {
        wB1a[s] = fB1[(s * 2 + 0) * 32 + lane];
        wB1b[s] = fB1[(s * 2 + 1) * 32 + lane];
    }
    v2f wB2a[8], wB2b[8];
    #pragma unroll
    for (int s = 0; s < 8; ++s) {
        wB2a[s] = fB2[(s * 2 + 0) * 32 + lane];
        wB2b[s] = fB2[(s * 2 + 1) * 32 + lane];
    }
    const float bia0 = sB1[lidx];
    const float bia1 = sB1[lidx + 16];
    const float bmi0 = sBm[lidx];
    const float bmi1 = sBm[lidx + 16];
    const float w20  = sW2[lidx];
    const float w21  = sW2[lidx + 16];

    // Per-lane A-fragment loader: 6 in-range K-pairs + value-masked tail (k=24).
    auto loadA = [&](int tile, v2f* areg) {
        const int row = tile * 16 + lidx;
        const float* Arow = history + ((size_t)row * D_SZ + d) * M_SZ;
        #pragma unroll
        for (int s = 0; s < 6; ++s) {
            const int kb = 4 * s + 2 * lhalf;         // max 23 < 25: always in range
            areg[s].x = Arow[kb];
            areg[s].y = Arow[kb + 1];
        }
        areg[6].x = Arow[24] * lmask;                 // unconditional in-bounds load, value-masked
        areg[6].y = 0.0f;
    };

    // 32 row-tiles of 16 batch rows; 8 waves x 4 tiles each.
    // Software-pipelined: tile j+1's history loads overlap tile j's GEMM2/GELU.
    v2f areg[7];
    loadA(wave, areg);

    #pragma unroll
    for (int j = 0; j < 4; ++j) {
        const int tile = wave + 8 * j;               // 0..31

        // ---- GEMM1: [16 x 28] x [28 x 32] via v_wmma_f32_16x16x4_f32 ----
        v8f c0 = {};
        v8f c1 = {};
        #pragma unroll
        for (int s = 0; s < 7; ++s) {
            c0 = __builtin_amdgcn_wmma_f32_16x16x4_f32(false, areg[s], false, wB1a[s], (short)0, c0, false, false);
            c1 = __builtin_amdgcn_wmma_f32_16x16x4_f32(false, areg[s], false, wB1b[s], (short)0, c1, false, false);
        }

        // ---- prefetch next tile's A-fragments (overlaps with GELU + GEMM2 below) ----
        if (j < 3) loadA(wave + 8 * (j + 1), areg);

        // ---- bias + GELU, park activation tile in per-wave LDS (row-major 16x32) ----
        #pragma unroll
        for (int i = 0; i < 8; ++i) {
            const int r = i + 8 * lhalf;             // C-layout: VGPR i holds rows i and i+8
            sHbuf[wave][r * HB_STRIDE + lidx]      = gelu_erf(c0[i] + bia0);
            sHbuf[wave][r * HB_STRIDE + 16 + lidx] = gelu_erf(c1[i] + bia1);
        }
        // no barrier needed: LDS ops are in-order within a wave and hbuf is wave-private

        // ---- GEMM2: [16 x 32] x [32 x 32] ----
        v8f e0 = {};
        v8f e1 = {};
        #pragma unroll
        for (int s = 0; s < 8; ++s) {
            const int kb = 4 * s + 2 * lhalf;        // even; row base even -> 8B-aligned pair
            const v2f a = *(const v2f*)&sHbuf[wave][lidx * HB_STRIDE + kb];
            e0 = __builtin_amdgcn_wmma_f32_16x16x4_f32(false, a, false, wB2a[s], (short)0, e0, false, false);
            e1 = __builtin_amdgcn_wmma_f32_16x16x4_f32(false, a, false, wB2b[s], (short)0, e1, false, false);
        }

        // ---- bias + GELU + per-row dot with W2 + 16-lane butterfly reduce + store ----
        #pragma unroll
        for (int i = 0; i < 8; ++i) {
            float h0 = gelu_erf(e0[i] + bmi0);
            float h1 = gelu_erf(e1[i] + bmi1);
            float v  = h0 * w20 + h1 * w21;          // partial over this lane's two columns
            v += __shfl_xor(v, 8, 32);               // reduce across the 16 columns
            v += __shfl_xor(v, 4, 32);
            v += __shfl_xor(v, 2, 32);
            v += __shfl_xor(v, 1, 32);
            if (lidx == 0) {
                const int m = tile * 16 + i + 8 * lhalf;   // batch row of this C-layout slot
                out[(size_t)m * D_SZ + d] = v + b2d;
            }
        }
    }
}

extern "C" void kernel_launch(void* const* d_in, const int* in_sizes, int n_in,
                              void* d_out, int out_size, void* d_ws, size_t ws_size,
                              hipStream_t stream) {
    const float* history = (const float*)d_in[0];
    const float* W1      = (const float*)d_in[1];
    const float* b1      = (const float*)d_in[2];
    const float* Wm      = (const float*)d_in[3];
    const float* bm      = (const float*)d_in[4];
    const float* W2      = (const float*)d_in[5];
    const float* b2      = (const float*)d_in[6];
    float* out = (float*)d_out;

    dim3 grid(D_SZ);
    dim3 block(256);
    hipLaunchKernelGGL(neuron_mlp_kernel, grid, block, 0, stream,
                       history, W1, b1, Wm, bm, W2, b2, out);
}